// Sen2LikeFusion_25039659336280
// MI455X (gfx1250) — compile-verified
//
#include <hip/hip_runtime.h>
#include <hip/hip_bf16.h>
#include <math.h>

// ---------------------------------------------------------------------------
// Sen2Like fusion, MI455X (gfx1250, wave32).
// Bandwidth-bound workload (~350MB traffic @ 23.3TB/s ~ 17us); the 3x3
// separable Gaussian PSF conv is executed on the WMMA pipe as two banded
// 16x16x16 f32 matmuls per 14x14 output tile (V_WMMA_F32_16X16X4_F32 x8).
// ---------------------------------------------------------------------------

typedef float v2f __attribute__((ext_vector_type(2)));
typedef float v8f __attribute__((ext_vector_type(8)));

#define NT   20
#define NHR  30
#define NC   4
#define HH   512          // HR height/width
#define LH   256          // LR height/width
#define TD   14           // output tile dim (14x14 out of a 16x16 halo patch)
#define TPD  37           // ceil(512/14)
#define TILES_PER_IMG (TPD*TPD)
#define ABOVE 10000

// workspace layout (bytes)
#define WS_RATE   0       // float[30]
#define WS_FI     128     // int[20]
#define WS_SI     256     // int[20]
#define WS_FV     384     // float[20]
#define WS_SV     512     // float[20]
#define WS_VALID  640     // int[20]

// ---------------------------------------------------------------------------
// Kernel 1: rate[j] = mean(hr_mask[j]); bool bytes are 0/1 so popcount of a
// packed dword counts true pixels. One block per HR frame.
// ---------------------------------------------------------------------------
__global__ __launch_bounds__(256) void rate_kernel(
    const unsigned char* __restrict__ hr_mask, float* __restrict__ rate)
{
    __shared__ int red[256];
    const int j = blockIdx.x;
    const unsigned int* p =
        (const unsigned int*)(hr_mask + (size_t)j * (HH * HH));
    int s = 0;
    const int tid = threadIdx.x;
    // 512*512 bytes = 65536 dwords; 256 per thread, coalesced.
    for (int k = 0; k < 256; ++k)
        s += __popc(p[tid + k * 256]);
    red[tid] = s;
    __syncthreads();
    for (int off = 128; off > 0; off >>= 1) {
        if (tid < off) red[tid] += red[tid + off];
        __syncthreads();
    }
    if (tid == 0) rate[j] = (float)red[0] / (float)(HH * HH);
}

// ---------------------------------------------------------------------------
// Kernel 2: temporal frame selection (first/second nearest preceding HR frame)
// ---------------------------------------------------------------------------
__global__ void select_kernel(const int* __restrict__ lr_doy,
                              const int* __restrict__ hr_doy,
                              char* __restrict__ ws)
{
    const float* rate = (const float*)(ws + WS_RATE);
    int*   fiA = (int*)(ws + WS_FI);
    int*   siA = (int*)(ws + WS_SI);
    float* fvA = (float*)(ws + WS_FV);
    float* svA = (float*)(ws + WS_SV);
    int*   vaA = (int*)(ws + WS_VALID);

    const int t = threadIdx.x;
    if (t >= NT) return;
    const int ld = lr_doy[t];

    int best = 0x7fffffff, bi = 0;
    for (int j = 0; j < NHR; ++j) {
        int d = ld - hr_doy[j];
        if (d < 0 || rate[j] > 0.5f) d = ABOVE;
        if (d < best) { best = d; bi = j; }      // argmin, first occurrence
    }
    int best2 = 0x7fffffff, bi2 = 0;
    for (int j = 0; j < NHR; ++j) {
        int d = ld - hr_doy[j];
        if (d < 0 || rate[j] > 0.5f) d = ABOVE;
        if (j == bi) d = ABOVE;
        if (d < best2) { best2 = d; bi2 = j; }
    }
    fiA[t] = bi;
    siA[t] = bi2;
    fvA[t] = (float)hr_doy[bi];
    svA[t] = (float)hr_doy[bi2];
    vaA[t] = (best < ABOVE && best2 < ABOVE) ? 1 : 0;
}

// ---------------------------------------------------------------------------
// Kernel 3: fused main kernel.
//   Per wave: one 14x14 output tile of one (t,c) image.
//   P  = 16x16 halo patch of predicted_hr (zero outside image == SAME pad)
//   M1 = Tv * P     (vertical 3-tap pass; banded constant Tv as A operand)
//   L  = M1 * Th    (horizontal 3-tap pass; banded constant Th as B operand)
//   final_hr = lr_up + pred - L   (valid for rows/cols 0..13 of the tile)
// ---------------------------------------------------------------------------
__device__ __forceinline__ float band_w(int d, float gC, float gS) {
    // 1D normalized Gaussian tap for offset d in {-1,0,1}, else 0
    return (d == 0) ? gC : ((d == 1 || d == -1) ? gS : 0.0f);
}

__global__ __launch_bounds__(256) void fuse_kernel(
    const float* __restrict__ lr_data,
    const float* __restrict__ hr_data,
    const int*   __restrict__ lr_doy,
    const char*  __restrict__ ws,
    float* __restrict__ out_final,
    float* __restrict__ out_lrup)
{
    __shared__ float PsAll[8][16 * 17];   // halo patch, stride 17 (bank-safe)
    __shared__ float M1All[8][16 * 17];   // vertical-pass result

    const int lane = threadIdx.x & 31;
    const int wave = threadIdx.x >> 5;
    float* P  = PsAll[wave];
    float* M1 = M1All[wave];

    const int tileId = blockIdx.x * 8 + wave;       // exactly 80*1369 tiles
    const int img = tileId / TILES_PER_IMG;
    const int tyx = tileId % TILES_PER_IMG;
    const int t = img >> 2;
    const int c = img & 3;
    const int ty = tyx / TPD;
    const int tx = tyx % TPD;
    const int y0 = ty * TD;
    const int x0 = tx * TD;

    const int*   fiA = (const int*)(ws + WS_FI);
    const int*   siA = (const int*)(ws + WS_SI);
    const float* fvA = (const float*)(ws + WS_FV);
    const float* svA = (const float*)(ws + WS_SV);
    const int fi = fiA[t];
    const int si = siA[t];
    const float fv = fvA[t];
    const float sv = svA[t];
    const float doyF = (float)lr_doy[t];
    const float dv = sv - fv;

    const float* g1p = hr_data + (size_t)(fi * NC + c) * (HH * HH);
    const float* g2p = hr_data + (size_t)(si * NC + c) * (HH * HH);

    // gfx1250 prefetch hint for the gathered HR tiles (global_prefetch_b8)
    {
        int py = y0 > 0 ? (y0 - 1) : 0;
        int px = x0 > 0 ? (x0 - 1) : 0;
        __builtin_prefetch(g1p + (size_t)py * HH + px, 0, 0);
        __builtin_prefetch(g2p + (size_t)py * HH + px, 0, 0);
    }

    // 1D normalized Gaussian taps (MTF(Nyquist)=0.4): sigma^2 = 2 ln(2.5)/pi^2
    const double SIG2 = 1.8325814637483102 /
                        (3.141592653589793 * 3.141592653589793);
    const double e1 = exp(-1.0 / (2.0 * SIG2));
    const double sm = 1.0 + 2.0 * e1;
    const float gC = (float)(1.0 / sm);
    const float gS = (float)(e1 / sm);

    // ---- fill P: predicted_hr over the 16x16 halo patch (zero-padded) ----
    for (int i = 0; i < 8; ++i) {
        const int idx = i * 32 + lane;            // 0..255
        const int r = idx >> 4;
        const int s = idx & 15;
        const int gy = y0 - 1 + r;
        const int gx = x0 - 1 + s;
        float pred = 0.0f;
        if (gy >= 0 && gy < HH && gx >= 0 && gx < HH) {
            const size_t o = (size_t)gy * HH + gx;
            const float a1 = g1p[o];
            const float a2 = g2p[o];
            const float ca = (a2 - a1) / dv;
            const float cb = a2 - ca * sv;
            pred = fmaf(ca, doyF, cb);
        }
        P[r * 17 + s] = pred;
    }
    __syncthreads();

    // fragment index helpers (ISA 7.12.2 layouts, wave32)
    const int halfsel = (lane >> 4) << 1;   // 0 for lanes 0-15, 2 for 16-31
    const int mn = lane & 15;               // A-frag row / B-frag col
    const int rowD = (lane < 16) ? 0 : 8;   // C/D frag row base

    // ---- pass 1: M1 = Tv * P  (vertical 3-tap), 4 x wmma_f32_16x16x4 ----
    v8f acc = {0.f, 0.f, 0.f, 0.f, 0.f, 0.f, 0.f, 0.f};
    for (int kc = 0; kc < 4; ++kc) {
        const int k0 = kc * 4 + halfsel;
        v2f a, b;
        a.x = band_w((k0 + 0) - 1 - mn, gC, gS);      // Tv[mn, k0]
        a.y = band_w((k0 + 1) - 1 - mn, gC, gS);      // Tv[mn, k0+1]
        b.x = P[(k0 + 0) * 17 + mn];                  // P[k0,   mn]
        b.y = P[(k0 + 1) * 17 + mn];                  // P[k0+1, mn]
        acc = __builtin_amdgcn_wmma_f32_16x16x4_f32(
            false, a, false, b, (short)0, acc, false, false);
    }
    // D-frag -> LDS (row = v + rowD, col = mn)
    for (int v = 0; v < 8; ++v)
        M1[(rowD + v) * 17 + mn] = acc[v];
    __syncthreads();

    // ---- pass 2: L = M1 * Th  (horizontal 3-tap) ----
    v8f acc2 = {0.f, 0.f, 0.f, 0.f, 0.f, 0.f, 0.f, 0.f};
    for (int kc = 0; kc < 4; ++kc) {
        const int k0 = kc * 4 + halfsel;
        v2f a, b;
        a.x = M1[mn * 17 + (k0 + 0)];                 // M1[mn, k0]
        a.y = M1[mn * 17 + (k0 + 1)];
        b.x = band_w((k0 + 0) - 1 - mn, gC, gS);      // Th[k0,   mn]
        b.y = band_w((k0 + 1) - 1 - mn, gC, gS);
        acc2 = __builtin_amdgcn_wmma_f32_16x16x4_f32(
            false, a, false, b, (short)0, acc2, false, false);
    }

    // ---- epilogue: bilinear lr_up, final_hr = lr_up + pred - low ----
    const float* lrp = lr_data + (size_t)(t * NC + c) * (LH * LH);
    const size_t obase = (size_t)(t * NC + c) * (HH * HH);

    for (int v = 0; v < 8; ++v) {
        const int m = rowD + v;
        const int n = mn;
        const int y = y0 + m;
        const int x = x0 + n;
        if (m < TD && n < TD && y < HH && x < HH) {
            const float low  = acc2[v];
            const float pred = P[(m + 1) * 17 + (n + 1)];

            // half-pixel bilinear 2x upsample (jax.image.resize 'linear')
            const int ky = y >> 1;
            const int kx = x >> 1;
            int iy0, iy1, ix0, ix1;
            float wy0, wx0;
            if (y & 1) { iy0 = ky; iy1 = (ky + 1 < LH) ? ky + 1 : LH - 1;
                         wy0 = 0.75f; }
            else       { iy0 = (ky > 0) ? ky - 1 : 0; iy1 = ky;
                         wy0 = 0.25f; }
            if (x & 1) { ix0 = kx; ix1 = (kx + 1 < LH) ? kx + 1 : LH - 1;
                         wx0 = 0.75f; }
            else       { ix0 = (kx > 0) ? kx - 1 : 0; ix1 = kx;
                         wx0 = 0.25f; }
            const float wy1 = 1.0f - wy0;
            const float wx1 = 1.0f - wx0;
            const float l00 = lrp[(size_t)iy0 * LH + ix0];
            const float l01 = lrp[(size_t)iy0 * LH + ix1];
            const float l10 = lrp[(size_t)iy1 * LH + ix0];
            const float l11 = lrp[(size_t)iy1 * LH + ix1];
            const float lrup = wy0 * (wx0 * l00 + wx1 * l01) +
                               wy1 * (wx0 * l10 + wx1 * l11);

            const size_t o = obase + (size_t)y * HH + x;
            out_final[o] = lrup + (pred - low);
            out_lrup[o]  = lrup;
        }
    }
}

// ---------------------------------------------------------------------------
// Kernel 4: final_mask = lr_mask_up | ~valid | hr_mask[fi] | hr_mask[si]
// (bool -> 1.0f / 0.0f in the float output buffer)
// ---------------------------------------------------------------------------
__global__ __launch_bounds__(256) void mask_kernel(
    const unsigned char* __restrict__ lr_mask,
    const unsigned char* __restrict__ hr_mask,
    const char* __restrict__ ws,
    float* __restrict__ outm)
{
    const int idx = blockIdx.x * 256 + threadIdx.x;
    if (idx >= NT * HH * HH) return;
    const int t   = idx / (HH * HH);
    const int rem = idx % (HH * HH);
    const int y = rem >> 9;
    const int x = rem & 511;

    const int* fiA = (const int*)(ws + WS_FI);
    const int* siA = (const int*)(ws + WS_SI);
    const int* vaA = (const int*)(ws + WS_VALID);

    const int fi = fiA[t];
    const int si = siA[t];
    const int va = vaA[t];

    const bool lrm = lr_mask[(size_t)t * (LH * LH) + (y >> 1) * LH + (x >> 1)] != 0;
    const bool m1  = hr_mask[(size_t)fi * (HH * HH) + rem] != 0;
    const bool m2  = hr_mask[(size_t)si * (HH * HH) + rem] != 0;
    const bool m = lrm || (!va) || m1 || m2;
    outm[idx] = m ? 1.0f : 0.0f;
}

// ---------------------------------------------------------------------------
extern "C" void kernel_launch(void* const* d_in, const int* in_sizes, int n_in,
                              void* d_out, int out_size, void* d_ws,
                              size_t ws_size, hipStream_t stream)
{
    (void)in_sizes; (void)n_in; (void)out_size; (void)ws_size;

    const float*         lr_data = (const float*)d_in[0];
    const float*         hr_data = (const float*)d_in[1];
    const int*           lr_doy  = (const int*)d_in[2];
    const int*           hr_doy  = (const int*)d_in[3];
    const unsigned char* lr_mask = (const unsigned char*)d_in[4];
    const unsigned char* hr_mask = (const unsigned char*)d_in[5];

    float* out       = (float*)d_out;
    float* out_final = out;                                   // 20*4*512*512
    float* out_mask  = out + (size_t)NT * NC * HH * HH;       // 20*512*512
    float* out_lrup  = out_mask + (size_t)NT * HH * HH;       // 20*4*512*512
    char*  ws        = (char*)d_ws;

    rate_kernel<<<NHR, 256, 0, stream>>>(hr_mask, (float*)(ws + WS_RATE));
    select_kernel<<<1, 32, 0, stream>>>(lr_doy, hr_doy, ws);

    const int total_tiles = NT * NC * TILES_PER_IMG;          // 109520
    fuse_kernel<<<total_tiles / 8, 256, 0, stream>>>(
        lr_data, hr_data, lr_doy, ws, out_final, out_lrup);

    const int nmask = NT * HH * HH;
    mask_kernel<<<(nmask + 255) / 256, 256, 0, stream>>>(
        lr_mask, hr_mask, ws, out_mask);
}